// PNLoss_27831388078892
// MI455X (gfx1250) — compile-verified
//
#include <hip/hip_runtime.h>
#include <math.h>

typedef float v2f __attribute__((ext_vector_type(2)));
typedef float v8f __attribute__((ext_vector_type(8)));

#define S_SUP 8192
#define N_Q   2048
#define D_DIM 128
#define C_CLS 256

// ---------------------------------------------------------------------------
// Kernel 1: per-class prototype sums, counts, and squared norms.
// One block per class; ys staged once into LDS; fully deterministic.
// ---------------------------------------------------------------------------
__global__ __launch_bounds__(128) void pn_protos(const float* __restrict__ xs,
                                                 const int*   __restrict__ ys,
                                                 float* __restrict__ mus,
                                                 float* __restrict__ mu_norm,
                                                 float* __restrict__ cls_cnt) {
    __shared__ int   lys[S_SUP];
    __shared__ float red[128];
    const int c = blockIdx.x;
    const int t = threadIdx.x;

    for (int i = t; i < S_SUP; i += 128) lys[i] = ys[i];
    __syncthreads();

    float acc = 0.f;
    int   cnt = 0;
    for (int s = 0; s < S_SUP; ++s) {
        if (lys[s] == c) {                       // uniform branch (scalar)
            acc += xs[(size_t)s * D_DIM + t];
            ++cnt;
        }
    }
    mus[c * D_DIM + t] = acc;

    red[t] = acc * acc;
    __syncthreads();
    for (int off = 64; off > 0; off >>= 1) {
        if (t < off) red[t] += red[t + off];
        __syncthreads();
    }
    if (t == 0) {
        mu_norm[c] = red[0];
        cls_cnt[c] = (float)cnt;
    }
}

// ---------------------------------------------------------------------------
// Kernel 2: fused G = xq @ mus^T (f32 WMMA 16x16x4) + leave-one-out logits
// + log-softmax + per-block partial loss.
// Grid: N/16 = 128 blocks, blockDim = 512 (16 waves).
// Wave w computes tile (rows m0..m0+15, cols w*16..w*16+15), then row w's
// softmax. All reductions in fixed order -> deterministic.
// ---------------------------------------------------------------------------
__global__ __launch_bounds__(512) void pn_gemm_loss(const float* __restrict__ xq,
                                                    const int*   __restrict__ yq,
                                                    const float* __restrict__ mus,
                                                    const float* __restrict__ mu_norm,
                                                    const float* __restrict__ cls_cnt,
                                                    float* __restrict__ partial) {
    __shared__ float Gt[16][C_CLS];
    __shared__ float xqn[16];
    __shared__ float rowloss[16];

    const int t    = threadIdx.x;
    const int wave = t >> 5;      // 0..15
    const int lane = t & 31;
    const int lr   = lane & 15;
    const int hi   = lane >> 4;
    const int m0   = blockIdx.x * 16;

    // ---- ||xq[n]||^2 : wave w reduces row m0+w ----
    {
        const float* xrow = xq + (size_t)(m0 + wave) * D_DIM;
        float p = 0.f;
        #pragma unroll
        for (int j = 0; j < D_DIM; j += 32) {
            float v = xrow[j + lane];
            p += v * v;
        }
        #pragma unroll
        for (int off = 16; off > 0; off >>= 1) p += __shfl_xor(p, off, 32);
        if (lane == 0) xqn[wave] = p;
    }

    // ---- WMMA f32 16x16x4 tile: G[m0+.. , c0+..] ----
    const int c0 = wave * 16;
    v8f acc = {};
    const float* arow = xq  + (size_t)(m0 + lr) * D_DIM;   // A: 16x4, M = lr
    const float* brow = mus + (size_t)(c0 + lr) * D_DIM;   // B = mus^T, N = lr
    for (int k = 0; k < D_DIM; k += 4) {
        v2f a, b;
        a.x = arow[k + 2 * hi];
        a.y = arow[k + 2 * hi + 1];
        b.x = brow[k + 2 * hi];
        b.y = brow[k + 2 * hi + 1];
        acc = __builtin_amdgcn_wmma_f32_16x16x4_f32(
            /*neg_a=*/false, a, /*neg_b=*/false, b,
            /*c_mod=*/(short)0, acc, /*reuse_a=*/false, /*reuse_b=*/false);
    }
    // C/D layout: VGPR j -> row (j + 8*hi), col lr
    #pragma unroll
    for (int j = 0; j < 8; ++j) {
        Gt[j + 8 * hi][c0 + lr] = acc[j];
    }
    __syncthreads();

    // ---- per-row leave-one-out logits + log-softmax: wave w -> row w ----
    {
        const int   r  = wave;
        const int   n  = m0 + r;
        const int   y  = yq[n];
        const float xn = xqn[r];

        float mx = -3.0e38f;
        float ly = 0.f;                       // logit of target class
        for (int c = lane; c < C_CLS; c += 32) {
            float g   = Gt[r][c];
            float cnt = cls_cnt[c];
            float mn2 = mu_norm[c];
            float dot = g;
            if (c == y) {                     // leave-one-out class
                cnt -= 1.f;
                dot  = g - xn;
                mn2  = mn2 - 2.f * g + xn;
            }
            float denom = fmaxf(cnt, 0.1f);
            float inv   = 1.f / denom;
            float logit = -0.5f * (xn - 2.f * dot * inv + mn2 * inv * inv);
            if (!(cnt > 0.1f)) logit = 0.f;   // mask empty classes
            Gt[r][c] = logit;
            mx = fmaxf(mx, logit);
            if (c == y) ly = logit;
        }
        #pragma unroll
        for (int off = 16; off > 0; off >>= 1) {
            mx = fmaxf(mx, __shfl_xor(mx, off, 32));
            ly += __shfl_xor(ly, off, 32);    // only one lane held nonzero ly
        }
        float se = 0.f;
        for (int c = lane; c < C_CLS; c += 32) se += expf(Gt[r][c] - mx);
        #pragma unroll
        for (int off = 16; off > 0; off >>= 1) se += __shfl_xor(se, off, 32);
        if (lane == 0) {
            rowloss[r] = -(ly - mx - logf(se));
        }
    }
    __syncthreads();

    if (t == 0) {
        float s = 0.f;
        #pragma unroll
        for (int r = 0; r < 16; ++r) s += rowloss[r];
        partial[blockIdx.x] = s;
    }
}

// ---------------------------------------------------------------------------
// Kernel 3: deterministic final reduction of 128 block partials -> mean loss.
// ---------------------------------------------------------------------------
__global__ __launch_bounds__(128) void pn_final(const float* __restrict__ partial,
                                                float* __restrict__ out) {
    __shared__ float red[128];
    const int t = threadIdx.x;
    red[t] = partial[t];
    __syncthreads();
    for (int off = 64; off > 0; off >>= 1) {
        if (t < off) red[t] += red[t + off];
        __syncthreads();
    }
    if (t == 0) out[0] = red[0] / (float)N_Q;
}

// ---------------------------------------------------------------------------
extern "C" void kernel_launch(void* const* d_in, const int* in_sizes, int n_in,
                              void* d_out, int out_size, void* d_ws, size_t ws_size,
                              hipStream_t stream) {
    const float* xq  = (const float*)d_in[0];   // [N, D]
    const int*   yq  = (const int*)  d_in[1];   // [N]
    const float* xs  = (const float*)d_in[2];   // [S, D]
    const int*   ys  = (const int*)  d_in[3];   // [S]
    // d_in[4] = pos, unused (yq already equals ys[pos])

    float* ws       = (float*)d_ws;
    float* mus      = ws;                        // C*D   = 32768 floats
    float* mu_norm  = ws + C_CLS * D_DIM;        // C     = 256
    float* cls_cnt  = mu_norm + C_CLS;           // C     = 256
    float* partial  = cls_cnt + C_CLS;           // N/16  = 128

    float* out = (float*)d_out;

    pn_protos   <<<C_CLS,    128, 0, stream>>>(xs, ys, mus, mu_norm, cls_cnt);
    pn_gemm_loss<<<N_Q / 16, 512, 0, stream>>>(xq, yq, mus, mu_norm, cls_cnt, partial);
    pn_final    <<<1,        128, 0, stream>>>(partial, out);
}